// CoreTransformerEncoder_23682449670146
// MI455X (gfx1250) — compile-verified
//
#include <hip/hip_runtime.h>
#include <math.h>

// CDNA5 / gfx1250, wave32.
typedef __attribute__((ext_vector_type(2))) float v2f;
typedef __attribute__((ext_vector_type(8))) float v8f;

#define D_MODEL 256
#define NSEQ    256
#define NHEAD   8
#define HEAD_DIM 32
#define BATCH   2
#define ROWS    (BATCH * NSEQ)   // 512

// ---------------- Kernel A: x = LN(nodes); nq = x @ W + b ----------------
__global__ __launch_bounds__(256) void prep_kernel(
    const float* __restrict__ nodes, const float* __restrict__ W,
    const float* __restrict__ bvec, const float* __restrict__ ln_g,
    const float* __restrict__ ln_b, float* __restrict__ xG,
    float* __restrict__ nqG)
{
    __shared__ float sX[256];
    __shared__ float sR[256];
    const int r = blockIdx.x;      // b*256 + i
    const int t = threadIdx.x;

    float v = nodes[(size_t)r * 256 + t];
    sR[t] = v;
    __syncthreads();
    for (int off = 128; off > 0; off >>= 1) {
        if (t < off) sR[t] += sR[t + off];
        __syncthreads();
    }
    float mu = sR[0] * (1.0f / 256.0f);
    __syncthreads();
    float d = v - mu;
    sR[t] = d * d;
    __syncthreads();
    for (int off = 128; off > 0; off >>= 1) {
        if (t < off) sR[t] += sR[t + off];
        __syncthreads();
    }
    float var = sR[0] * (1.0f / 256.0f);
    float rs = rsqrtf(var + 1e-5f);
    float x = d * rs * ln_g[t] + ln_b[t];
    sX[t] = x;
    xG[(size_t)r * 256 + t] = x;
    __syncthreads();

    float acc = bvec[t];
    #pragma unroll 8
    for (int k = 0; k < 256; ++k)
        acc += sX[k] * W[(size_t)k * 256 + t];
    nqG[(size_t)r * 256 + t] = acc;
}

// ---------------- Kernel B: fused attention + FFN per query row ----------------
// Dynamic LDS layout (floats):
#define SA_F   (64 * 68)    // edges tile 64 rows x 64 k, padded to 68
#define SB_F   (64 * 260)   // W tile 64 x 256, padded to 260 (aliased by sEq 64x260)
#define SS_F   (8 * 256)    // scores / attn  [h][j]
#define TOTAL_SMEM_F (SA_F + SB_F + SS_F + 256 + 256 + 8 + 8 + 256 + 256 + 1024 + 256)

__global__ __launch_bounds__(512) void attn_kernel(
    const float* __restrict__ edges, const float* __restrict__ W,
    const float* __restrict__ bvec, const float* __restrict__ ln_g,
    const float* __restrict__ ln_b, const float* __restrict__ W1,
    const float* __restrict__ b1, const float* __restrict__ W2,
    const float* __restrict__ b2, const float* __restrict__ xG,
    const float* __restrict__ nqG, float* __restrict__ outG)
{
    extern __shared__ float smem[];
    float* sA   = smem;                 // [64][68]
    float* sB   = sA + SA_F;            // [64][260]  (W chunk)
    float* sEq  = sB;                   // [64][260]  ALIAS: eq tile, live only between barriers
    float* sS   = sB + SB_F;            // [8][256]
    float* sQi  = sS + SS_F;            // [256]
    float* sRed = sQi + 256;            // [256]
    float* sMx  = sRed + 256;           // [8]
    float* sInv = sMx + 8;              // [8]
    float* sOut = sInv + 8;             // [256]
    float* sZ   = sOut + 256;           // [256]
    float* sH   = sZ + 256;             // [1024]
    float* sTmp = sH + 1024;            // [256]

    const int tid  = threadIdx.x;
    const int lane = tid & 31;
    const int wv   = tid >> 5;                 // wave 0..15
    const int bi   = blockIdx.x;               // b*256 + i
    const int b    = bi >> 8;
    const size_t edgeRowBase = (size_t)bi * 256 * 256;  // edges[(bi*256+j)*256+k]

    const int n0      = wv << 4;               // this wave's output column tile
    const int lane_lo = lane & 15;
    const int khalf   = (lane >> 4) << 1;      // K sub-offset per ISA A/B layout
    const float biasN = bvec[n0 + lane_lo];

    if (tid < 256) sQi[tid] = nqG[(size_t)bi * 256 + tid];
    __syncthreads();

    // ---- streaming pass over j in 64-row groups (4 j-tiles per wave) ----
    for (int jg = 0; jg < 4; ++jg) {
        const int j0 = jg << 6;
        v8f acc[4] = {{}, {}, {}, {}};
        for (int k0 = 0; k0 < 256; k0 += 64) {
            __syncthreads();  // protects sA/sB (incl. sEq alias of previous group)
            // edges tile: 64 rows x 64 cols = 1024 float4, 2 per thread
            {
                int idx = tid;
                #pragma unroll
                for (int it = 0; it < 2; ++it, idx += 512) {
                    int r = idx >> 4, c4 = (idx & 15) << 2;
                    float4 v = *(const float4*)&edges[edgeRowBase + (size_t)(j0 + r) * 256 + (k0 + c4)];
                    *(float4*)&sA[r * 68 + c4] = v;
                }
            }
            // W tile: 64 rows x 256 cols = 4096 float4, 8 per thread
            #pragma unroll
            for (int it = 0; it < 8; ++it) {
                int idx = tid + (it << 9);
                int r = idx >> 6, c4 = (idx & 63) << 2;
                float4 v = *(const float4*)&W[(size_t)(k0 + r) * 256 + c4];
                *(float4*)&sB[r * 260 + c4] = v;
            }
            // prefetch next k-chunk of edges (speculative, stays in-bounds)
            if (k0 < 192 && tid < 64)
                __builtin_prefetch(&edges[edgeRowBase + (size_t)(j0 + tid) * 256 + (k0 + 64)], 0, 0);
            __syncthreads();
            #pragma unroll
            for (int kk = 0; kk < 16; ++kk) {
                int kb = (kk << 2) + khalf;
                v2f bf;                             // B fragment, reused for 4 j-tiles
                bf.x = sB[kb * 260 + n0 + lane_lo];
                bf.y = sB[(kb + 1) * 260 + n0 + lane_lo];
                #pragma unroll
                for (int jt = 0; jt < 4; ++jt) {
                    v2f a;
                    a.x = sA[(jt * 16 + lane_lo) * 68 + kb];
                    a.y = sA[(jt * 16 + lane_lo) * 68 + kb + 1];
                    acc[jt] = __builtin_amdgcn_wmma_f32_16x16x4_f32(
                        false, a, false, bf, (short)0, acc[jt], false, false);
                }
            }
        }
        __syncthreads();   // all waves done reading sB -> safe to overwrite via sEq alias
        // write 64x16 tile (+bias) to sEq per C/D VGPR layout
        {
            int mbase = (lane >> 4) << 3;
            #pragma unroll
            for (int jt = 0; jt < 4; ++jt) {
                #pragma unroll
                for (int r = 0; r < 8; ++r)
                    sEq[(jt * 16 + mbase + r) * 260 + n0 + lane_lo] = acc[jt][r] + biasN;
            }
        }
        __syncthreads();
        // scores: 512 threads = 64 j x 8 h;  s(j,h) = sum_c (e+qi)(e+qj)
        {
            int j = tid >> 3, h = tid & 7, cb = h << 5;
            int jgl = j0 + j;
            const float* qj = &nqG[((size_t)(b << 8) + jgl) * 256 + cb];
            float s = 0.f;
            #pragma unroll
            for (int c = 0; c < 32; ++c) {
                float e = sEq[j * 260 + cb + c];
                s += (e + sQi[cb + c]) * (e + qj[c]);
            }
            s *= 0.17677669529663689f;          // 1/sqrt(32)
            s = 10.0f * tanhf(s);               // CLIP * tanh
            sS[h * 256 + jgl] = s;
        }
    }
    __syncthreads();

    // ---- softmax over j, per head ----
    if (tid < 256) {
        int h = tid >> 5, l = tid & 31;
        float m = -1e30f;
        for (int j = l; j < 256; j += 32) m = fmaxf(m, sS[h * 256 + j]);
        sRed[tid] = m;
    }
    __syncthreads();
    if (tid < 8) {
        float m = -1e30f;
        for (int l = 0; l < 32; ++l) m = fmaxf(m, sRed[tid * 32 + l]);
        sMx[tid] = m;
    }
    __syncthreads();
    if (tid < 256) {
        int h = tid >> 5, l = tid & 31;
        float m = sMx[h], sum = 0.f;
        for (int j = l; j < 256; j += 32) {
            float e = expf(sS[h * 256 + j] - m);
            sS[h * 256 + j] = e;
            sum += e;
        }
        sRed[tid] = sum;
    }
    __syncthreads();
    if (tid < 8) {
        float s = 0.f;
        for (int l = 0; l < 32; ++l) s += sRed[tid * 32 + l];
        sInv[tid] = 1.0f / s;
    }
    __syncthreads();

    // ---- out[c] = sum_j attn[h,j] * v[j,c]   (v == nq, shared weights) ----
    if (tid < 256) {
        int c = tid, h = c >> 5;
        float acc2 = 0.f;
        const float* vb = &nqG[(size_t)(b << 8) * 256 + c];
        for (int j = 0; j < 256; ++j) acc2 += sS[h * 256 + j] * vb[(size_t)j * 256];
        sOut[c] = acc2 * sInv[h];
    }
    __syncthreads();

    // ---- output projection + residual ----
    if (tid < 256) {
        int c = tid;
        float acc2 = bvec[c];
        #pragma unroll 8
        for (int k = 0; k < 256; ++k) acc2 += sOut[k] * W[(size_t)k * 256 + c];
        sTmp[c] = xG[(size_t)bi * 256 + c] + acc2;
    }
    __syncthreads();

    // ---- LayerNorm of residual ----
    if (tid < 256) sRed[tid] = sTmp[tid];
    __syncthreads();
    for (int off = 128; off > 0; off >>= 1) {
        if (tid < off) sRed[tid] += sRed[tid + off];
        __syncthreads();
    }
    float mu = sRed[0] * (1.0f / 256.0f);
    __syncthreads();
    if (tid < 256) { float d = sTmp[tid] - mu; sRed[tid] = d * d; }
    __syncthreads();
    for (int off = 128; off > 0; off >>= 1) {
        if (tid < off) sRed[tid] += sRed[tid + off];
        __syncthreads();
    }
    float var = sRed[0] * (1.0f / 256.0f);
    float rs = rsqrtf(var + 1e-5f);
    __syncthreads();
    if (tid < 256) sZ[tid] = (sTmp[tid] - mu) * rs * ln_g[tid] + ln_b[tid];
    __syncthreads();

    // ---- FFN: gelu(z@W1+b1)@W2+b2 + z ----
    #pragma unroll
    for (int q = 0; q < 2; ++q) {
        int col = tid + (q << 9);   // 512 threads cover 1024 hidden cols
        float acc2 = b1[col];
        #pragma unroll 8
        for (int k = 0; k < 256; ++k) acc2 += sZ[k] * W1[(size_t)k * 1024 + col];
        acc2 = 0.5f * acc2 * (1.0f + erff(acc2 * 0.70710678118654752f));  // exact gelu
        sH[col] = acc2;
    }
    __syncthreads();
    if (tid < 256) {
        int c = tid;
        float acc2 = b2[c];
        #pragma unroll 8
        for (int k = 0; k < 1024; ++k) acc2 += sH[k] * W2[(size_t)k * 256 + c];
        outG[(size_t)bi * 256 + c] = acc2 + sZ[c];
    }
}

extern "C" void kernel_launch(void* const* d_in, const int* in_sizes, int n_in,
                              void* d_out, int out_size, void* d_ws, size_t ws_size,
                              hipStream_t stream) {
    const float* nodes = (const float*)d_in[0];
    const float* edges = (const float*)d_in[1];
    const float* W     = (const float*)d_in[2];
    const float* bv    = (const float*)d_in[3];
    const float* g     = (const float*)d_in[4];
    const float* lb    = (const float*)d_in[5];
    const float* W1    = (const float*)d_in[6];
    const float* b1    = (const float*)d_in[7];
    const float* W2    = (const float*)d_in[8];
    const float* b2    = (const float*)d_in[9];
    float* out = (float*)d_out;

    // workspace: x (post-LN nodes) 512x256, nq 512x256  -> 1 MB
    float* xG  = (float*)d_ws;
    float* nqG = xG + (size_t)ROWS * D_MODEL;

    prep_kernel<<<ROWS, 256, 0, stream>>>(nodes, W, bv, g, lb, xG, nqG);

    size_t shmem = (size_t)TOTAL_SMEM_F * sizeof(float);   // ~101 KB (< 160 KB CU-mode cap)
    attn_kernel<<<ROWS, 512, shmem, stream>>>(edges, W, bv, g, lb, W1, b1, W2, b2,
                                              xG, nqG, out);
}